// Eikonal_10943576670372
// MI455X (gfx1250) — compile-verified
//
#include <hip/hip_runtime.h>
#include <math.h>

#define TPB 256
#define NB_ROWS 8   // B in the reference

// clang ext-vector types: __builtin_nontemporal_load/store accept these.
typedef int   v4i __attribute__((ext_vector_type(4)));
typedef float v4f __attribute__((ext_vector_type(4)));

// ---------------------------------------------------------------------------
// Kernel 1: zero the degree array and the transposed accumulator.
// ---------------------------------------------------------------------------
__global__ void eik_init_ws(float* __restrict__ deg, float* __restrict__ accT,
                            int N, int NB) {
  int i = blockIdx.x * TPB + threadIdx.x;
  if (i < NB) accT[i] = 0.0f;
  if (i < N)  deg[i]  = 0.0f;
}

// ---------------------------------------------------------------------------
// Kernel 2: deg = segment_sum(graph_edge_attr, ei0).
// 4 edges/thread, non-temporal b128 loads of the use-once streams,
// native non-returning global_atomic_add_f32 into the L2-resident deg[].
// ---------------------------------------------------------------------------
__global__ void eik_deg(const int* __restrict__ ei0,
                        const float* __restrict__ ea,
                        float* __restrict__ deg, int E) {
  int e4 = (blockIdx.x * TPB + threadIdx.x) * 4;
  if (e4 >= E) return;
  __builtin_prefetch(ei0 + e4 + 16384, 0, 1);   // global_prefetch_b8
  __builtin_prefetch(ea  + e4 + 16384, 0, 1);
  if (e4 + 3 < E) {
    v4i i0 = __builtin_nontemporal_load((const v4i*)(ei0 + e4));
    v4f a  = __builtin_nontemporal_load((const v4f*)(ea  + e4));
#pragma unroll
    for (int k = 0; k < 4; ++k)
      unsafeAtomicAdd(&deg[i0[k]], a[k]);
  } else {
    for (int k = 0; k < 4 && e4 + k < E; ++k)
      unsafeAtomicAdd(&deg[ei0[e4 + k]], ea[e4 + k]);
  }
}

// ---------------------------------------------------------------------------
// Kernel 3: yT[n][b] = y[b][n]  (makes per-edge gathers two b128 loads),
//           and deg[n] <- 1 / deg[n]^2  (ALPHA = 2), computed once per node.
// ---------------------------------------------------------------------------
__global__ void eik_transpose_invdeg(const float* __restrict__ y,
                                     float* __restrict__ yT,
                                     float* __restrict__ deg, int N) {
  int i = blockIdx.x * TPB + threadIdx.x;
  if (i >= N) return;
  float d = deg[i];
  deg[i] = 1.0f / (d * d);
#pragma unroll
  for (int b = 0; b < NB_ROWS; ++b)
    yT[(size_t)i * NB_ROWS + b] = y[(size_t)b * N + i];
}

// ---------------------------------------------------------------------------
// Kernel 4: per-edge gradient + scatter-add into accT[N][8].
//   diff[b] = y[b,ei1]*inv - y[b,ei0]*inv,  inv = 1/deg[ei0]^2
//   g[b]    = sqrt(sigmoid(dist_w)) * max(-diff[b], 0)   (== |grad_m|)
// 4 edges/thread; streams are NT b128 loads; yT gathers are RT b128 loads
// (L2-resident); atomics predicated on g>0 (EXEC-masked, ~2x less traffic).
// ---------------------------------------------------------------------------
__global__ void eik_edges(const int*   __restrict__ ei0,
                          const int*   __restrict__ ei1,
                          const float* __restrict__ dist_w,
                          const float* __restrict__ invdeg2,
                          const v4f*   __restrict__ yT,      // [N][2] v4f
                          float* __restrict__ accT,          // [N][8]
                          int E) {
  int e4 = (blockIdx.x * TPB + threadIdx.x) * 4;
  if (e4 >= E) return;

  __builtin_prefetch(ei0    + e4 + 16384, 0, 1);
  __builtin_prefetch(ei1    + e4 + 16384, 0, 1);
  __builtin_prefetch(dist_w + e4 + 16384, 0, 1);

  v4i i0v, i1v;
  v4f dwv;
  int nvalid;
  if (e4 + 3 < E) {
    i0v = __builtin_nontemporal_load((const v4i*)(ei0    + e4));
    i1v = __builtin_nontemporal_load((const v4i*)(ei1    + e4));
    dwv = __builtin_nontemporal_load((const v4f*)(dist_w + e4));
    nvalid = 4;
  } else {
    nvalid = E - e4;
    for (int k = 0; k < nvalid; ++k) {
      i0v[k] = ei0[e4 + k];
      i1v[k] = ei1[e4 + k];
      dwv[k] = dist_w[e4 + k];
    }
  }

  for (int k = 0; k < nvalid; ++k) {
    int   i0  = i0v[k];
    int   i1  = i1v[k];
    float inv = invdeg2[i0];                        // L2-resident gather
    float s   = sqrtf(1.0f / (1.0f + expf(-dwv[k]))); // sqrt(sigmoid)

    // Two b128 gathers per endpoint (8 floats each), L2-resident.
    v4f a0 = yT[(size_t)i0 * 2 + 0];
    v4f a1 = yT[(size_t)i0 * 2 + 1];
    v4f b0 = yT[(size_t)i1 * 2 + 0];
    v4f b1 = yT[(size_t)i1 * 2 + 1];

    float* acc = accT + (size_t)i0 * NB_ROWS;
#pragma unroll
    for (int j = 0; j < 4; ++j) {
      float g = s * fmaxf(a0[j] * inv - b0[j] * inv, 0.0f);
      if (g > 0.0f) unsafeAtomicAdd(acc + j, g);
    }
#pragma unroll
    for (int j = 0; j < 4; ++j) {
      float g = s * fmaxf(a1[j] * inv - b1[j] * inv, 0.0f);
      if (g > 0.0f) unsafeAtomicAdd(acc + 4 + j, g);
    }
  }
}

// ---------------------------------------------------------------------------
// Kernel 5: out[b][n] = 1 - accT[n][b]
// b128 RT reads of the accumulator, non-temporal coalesced b32 writes
// (output is written once and never re-read on device).
// ---------------------------------------------------------------------------
__global__ void eik_finalize(const v4f* __restrict__ accT,
                             float* __restrict__ out, int N) {
  int i = blockIdx.x * TPB + threadIdx.x;
  if (i >= N) return;
  v4f a0 = accT[(size_t)i * 2 + 0];
  v4f a1 = accT[(size_t)i * 2 + 1];
#pragma unroll
  for (int b = 0; b < 4; ++b)
    __builtin_nontemporal_store(1.0f - a0[b], out + (size_t)b * N + i);
#pragma unroll
  for (int b = 0; b < 4; ++b)
    __builtin_nontemporal_store(1.0f - a1[b], out + (size_t)(b + 4) * N + i);
}

// ---------------------------------------------------------------------------
extern "C" void kernel_launch(void* const* d_in, const int* in_sizes, int n_in,
                              void* d_out, int out_size, void* d_ws, size_t ws_size,
                              hipStream_t stream) {
  const float* y    = (const float*)d_in[0];   // [B,N] f32
  // d_in[1] = t (unused)
  const float* ea   = (const float*)d_in[2];   // [E]   f32
  const float* dw   = (const float*)d_in[3];   // [E]   f32
  const int*   eidx = (const int*)  d_in[4];   // [2,E] int

  const int N  = in_sizes[0] / NB_ROWS;        // y is [8, N]
  const int E  = in_sizes[2];
  const int NB = N * NB_ROWS;

  const int* ei0 = eidx;
  const int* ei1 = eidx + E;

  // Workspace layout (floats): deg[N] | yT[N*8] | accT[N*8]  (~6.8 MB)
  float* deg  = (float*)d_ws;
  float* yT   = deg + N;
  float* accT = yT + (size_t)NB;
  float* out  = (float*)d_out;

  const int E4 = (E + 3) / 4;                  // threads for 4-edge kernels
  dim3 blk(TPB);
  eik_init_ws         <<<(NB + TPB - 1) / TPB, blk, 0, stream>>>(deg, accT, N, NB);
  eik_deg             <<<(E4 + TPB - 1) / TPB, blk, 0, stream>>>(ei0, ea, deg, E);
  eik_transpose_invdeg<<<(N  + TPB - 1) / TPB, blk, 0, stream>>>(y, yT, deg, N);
  eik_edges           <<<(E4 + TPB - 1) / TPB, blk, 0, stream>>>(
      ei0, ei1, dw, deg, (const v4f*)yT, accT, E);
  eik_finalize        <<<(N  + TPB - 1) / TPB, blk, 0, stream>>>(
      (const v4f*)accT, out, N);
}